// RiemannianBatchNorm_80642305950523
// MI455X (gfx1250) — compile-verified
//
#include <hip/hip_runtime.h>
#include <math.h>

// RiemannianBatchNorm for 8192 x (64x64) SPD matrices on gfx1250.
// - All spectral functions via matmul-only iterations -> v_wmma_f32_16x16x4_f32
// - Inputs double-buffered into LDS with global_load_async_to_lds_b128 (ASYNCcnt)

typedef __attribute__((ext_vector_type(2))) float v2f;
typedef __attribute__((ext_vector_type(8))) float v8f;

#define TPB 128            // 4 wave32 per block
#define LDA 68             // padded LDS row stride: 272B = 16B-aligned, low bank conflicts
#define MSZ (64 * LDA)     // floats per padded 64x64 LDS matrix
#define NS_ITERS 12        // Newton-Schulz iterations per sqrt
#define LOG_SQRTS 3        // inverse scaling-and-squaring depth
#define LOG_TERMS 8        // Mercator series terms
#define EXP_SQ 5           // exp squarings (scale 1/32)

__device__ __forceinline__ int pidx(int e) { return (e >> 6) * LDA + (e & 63); }

// ---- gfx1250 async copy: global -> LDS, bypassing VGPRs (tracked by ASYNCcnt) ----
// Copies one dense 64x64 f32 matrix from global into a padded LDS matrix.
// 1024 16B chunks; 8 per thread. Chunks never cross a row (64-float rows, 4-aligned).
__device__ __forceinline__ void async_load_mat(const float* __restrict__ gsrc,
                                               float* lds_dst) {
  // Generic LDS pointer low 32 bits == LDS byte offset (aperture base is 4GB-aligned).
  unsigned base = (unsigned)(unsigned long long)lds_dst;
  const int t = threadIdx.x;
#pragma unroll
  for (int i = 0; i < 8; ++i) {
    int q = t + i * TPB;          // chunk 0..1023
    int e = q * 4;                // first element of the 4-float chunk
    unsigned loff = base + (unsigned)pidx(e) * 4u;
    const float* g = gsrc + e;
    asm volatile("global_load_async_to_lds_b128 %0, %1, off"
                 :: "v"(loff), "v"(g)
                 : "memory");
  }
}

__device__ __forceinline__ void wait_async() {
  asm volatile("s_wait_asynccnt 0" ::: "memory");
}

// C = A @ B, 64x64 matrices in LDS with row stride LDA. C must not alias A/B.
// Block = 128 threads. Ends with a block barrier.
__device__ __forceinline__ void mm64(const float* A, const float* Bm, float* C) {
  const int lane = threadIdx.x & 31;
  const int wave = threadIdx.x >> 5;
  const int half = lane >> 4;
  const int l16  = lane & 15;
  v8f zero = {0.f, 0.f, 0.f, 0.f, 0.f, 0.f, 0.f, 0.f};
  v8f acc[4];
  acc[0] = zero; acc[1] = zero; acc[2] = zero; acc[3] = zero;
  const float* arow = A + (wave * 16 + l16) * LDA + half * 2;
#pragma unroll 4
  for (int k0 = 0; k0 < 64; k0 += 4) {
    v2f a;
    a.x = arow[k0];
    a.y = arow[k0 + 1];
    const float* b0 = Bm + (k0 + half * 2) * LDA + l16;
#pragma unroll
    for (int tn = 0; tn < 4; ++tn) {
      v2f b;
      b.x = b0[tn * 16];
      b.y = b0[LDA + tn * 16];
      acc[tn] = __builtin_amdgcn_wmma_f32_16x16x4_f32(
          false, a, false, b, (short)0, acc[tn], false, false);
    }
  }
  float* crow = C + (wave * 16 + half * 8) * LDA + l16;
#pragma unroll
  for (int tn = 0; tn < 4; ++tn) {
#pragma unroll
    for (int r = 0; r < 8; ++r) crow[r * LDA + tn * 16] = acc[tn][r];
  }
  __syncthreads();
}

__device__ __forceinline__ float block_reduce_sum(float v, float* red) {
  const int t = threadIdx.x;
  red[t] = v;
  __syncthreads();
#pragma unroll
  for (int k = TPB / 2; k > 0; k >>= 1) {
    if (t < k) red[t] += red[t + k];
    __syncthreads();
  }
  float r = red[0];
  __syncthreads();
  return r;
}

// Coupled Newton-Schulz: returns Y ~ M^{1/2}, Z ~ M^{-1/2}. Clobbers all 4 buffers.
__device__ __forceinline__ void sqrt_both(float* M, float* t1, float* t2, float* t3,
                                          float* red, float** Yout, float** Zout) {
  float ss = 0.f;
  for (int e = threadIdx.x; e < 4096; e += TPB) { float v = M[pidx(e)]; ss += v * v; }
  float c = sqrtf(block_reduce_sum(ss, red));
  float invc = 1.0f / c;
  for (int e = threadIdx.x; e < 4096; e += TPB) {
    int i = pidx(e);
    M[i] *= invc;
    t1[i] = ((e >> 6) == (e & 63)) ? 1.0f : 0.0f;
  }
  __syncthreads();
  float *Y = M, *Z = t1, *T = t2, *W = t3;
#pragma unroll 1
  for (int it = 0; it < NS_ITERS; ++it) {
    mm64(Z, Y, T);  // T = Z@Y
    for (int e = threadIdx.x; e < 4096; e += TPB) {
      int i = pidx(e);
      float d = ((e >> 6) == (e & 63)) ? 1.5f : 0.0f;
      T[i] = d - 0.5f * T[i];
    }
    __syncthreads();
    mm64(Y, T, W);  // new Y
    mm64(T, Z, Y);  // new Z written into old Y storage
    float* o = Y; Y = W; W = Z; Z = o;
  }
  float sc = sqrtf(c), isc = 1.0f / sc;
  for (int e = threadIdx.x; e < 4096; e += TPB) {
    int i = pidx(e);
    Y[i] *= sc;
    Z[i] *= isc;
  }
  __syncthreads();
  *Yout = Y; *Zout = Z;
}

// log(M) via LOG_SQRTS repeated sqrts + Mercator series. Clobbers all buffers.
__device__ __forceinline__ float* log64(float* p0, float* p1, float* p2, float* p3,
                                        float* red) {
#pragma unroll 1
  for (int k = 0; k < LOG_SQRTS; ++k) {
    float *Y, *Z;
    sqrt_both(p0, p1, p2, p3, red, &Y, &Z);
    float* all[4] = {p0, p1, p2, p3};
    float* rem[3];
    int n = 0;
    for (int i = 0; i < 4; ++i)
      if (all[i] != Y) rem[n++] = all[i];
    p0 = Y; p1 = rem[0]; p2 = rem[1]; p3 = rem[2];
  }
  // E = M - I (in place in p0)
  for (int e = threadIdx.x; e < 4096; e += TPB)
    if ((e >> 6) == (e & 63)) p0[pidx(e)] -= 1.0f;
  __syncthreads();
  // Horner: log(I+E) = E (1/1 - E(1/2 - E(1/3 - ... )))
  float* P = p1; float* W = p2;
  for (int e = threadIdx.x; e < 4096; e += TPB)
    P[pidx(e)] = ((e >> 6) == (e & 63)) ? (1.0f / (float)LOG_TERMS) : 0.0f;
  __syncthreads();
#pragma unroll 1
  for (int j = LOG_TERMS - 1; j >= 1; --j) {
    mm64(p0, P, W);
    float aj = 1.0f / (float)j;
    for (int e = threadIdx.x; e < 4096; e += TPB) {
      int i = pidx(e);
      float d = ((e >> 6) == (e & 63)) ? aj : 0.0f;
      W[i] = d - W[i];
    }
    __syncthreads();
    float* t = P; P = W; W = t;
  }
  mm64(p0, P, p3);
  const float m = (float)(1 << LOG_SQRTS);
  for (int e = threadIdx.x; e < 4096; e += TPB) p3[pidx(e)] *= m;
  __syncthreads();
  return p3;
}

// exp(pref*T) via scaling-and-squaring. Clobbers T, a, b; result in returned buffer.
__device__ __forceinline__ float* exp64(float* T, float* a, float* b, float pref) {
  const float s0 = pref * (1.0f / 32.0f);
  for (int e = threadIdx.x; e < 4096; e += TPB) T[pidx(e)] *= s0;
  __syncthreads();
  float* Q = a; float* W = b;
  for (int e = threadIdx.x; e < 4096; e += TPB)
    Q[pidx(e)] = ((e >> 6) == (e & 63)) ? 1.0f : 0.0f;
  __syncthreads();
#pragma unroll 1
  for (int j = 8; j >= 1; --j) {
    mm64(T, Q, W);
    float aj = 1.0f / (float)j;
    for (int e = threadIdx.x; e < 4096; e += TPB) {
      int i = pidx(e);
      float d = ((e >> 6) == (e & 63)) ? 1.0f : 0.0f;
      W[i] = d + aj * W[i];
    }
    __syncthreads();
    float* t = Q; Q = W; W = t;
  }
#pragma unroll 1
  for (int k = 0; k < EXP_SQ; ++k) {
    mm64(Q, Q, W);
    float* t = Q; Q = W; W = t;
  }
  return Q;
}

// ---------------- Stage kernels ----------------

// Deterministic two-stage batch mean (no float atomics).
__global__ void k_mean_part(const float* __restrict__ x, float* __restrict__ part,
                            int Bn, int nchunk) {
  int e = (blockIdx.x & 15) * 256 + threadIdx.x;   // element 0..4095
  int c = blockIdx.x >> 4;                         // chunk 0..nchunk-1
  int cl = Bn / nchunk;
  float s = 0.f;
  int b0 = c * cl;
  for (int b = 0; b < cl; ++b) s += x[(size_t)(b0 + b) * 4096 + e];
  part[(size_t)c * 4096 + e] = s;
}

__global__ void k_mean_final(const float* __restrict__ part, float* __restrict__ mean,
                             int Bn, int nchunk) {
  int e = blockIdx.x * 256 + threadIdx.x;
  float s = 0.f;
  for (int c = 0; c < nchunk; ++c) s += part[(size_t)c * 4096 + e];
  mean[e] = s / (float)Bn;
}

// s = mean^{1/2}, si = mean^{-1/2}
__global__ __launch_bounds__(TPB) void k_prep1(const float* __restrict__ mean,
                                               float* __restrict__ s_out,
                                               float* __restrict__ si_out) {
  extern __shared__ float sm[];
  float* B0 = sm; float* B1 = sm + MSZ; float* B2 = sm + 2 * MSZ; float* B3 = sm + 3 * MSZ;
  float* red = sm + 4 * MSZ;
  for (int e = threadIdx.x; e < 4096; e += TPB) B0[pidx(e)] = mean[e];
  __syncthreads();
  float *Y, *Z;
  sqrt_both(B0, B1, B2, B3, red, &Y, &Z);
  for (int e = threadIdx.x; e < 4096; e += TPB) {
    s_out[e]  = Y[pidx(e)];
    si_out[e] = Z[pidx(e)];
  }
}

// tpart[blk] = sum over its batch slice of log(si x si); double-buffered async input
__global__ __launch_bounds__(TPB) void k_pass1(const float* __restrict__ x,
                                               const float* __restrict__ si,
                                               float* __restrict__ tpart, int Bn) {
  extern __shared__ float sm[];
  float* SI = sm;           float* ACC = sm + MSZ;     float* X0 = sm + 2 * MSZ;
  float* X1 = sm + 3 * MSZ; float* M   = sm + 4 * MSZ; float* t1 = sm + 5 * MSZ;
  float* t2 = sm + 6 * MSZ; float* t3  = sm + 7 * MSZ; float* red = sm + 8 * MSZ;
  for (int e = threadIdx.x; e < 4096; e += TPB) {
    SI[pidx(e)]  = si[e];
    ACC[pidx(e)] = 0.0f;
  }
  int b0 = (int)blockIdx.x;
  if (b0 < Bn) async_load_mat(x + (size_t)b0 * 4096, X0);
  float* Xc = X0; float* Xn = X1;
#pragma unroll 1
  for (int b = b0; b < Bn; b += (int)gridDim.x) {
    wait_async();
    __syncthreads();
    int nb = b + (int)gridDim.x;
    if (nb < Bn) async_load_mat(x + (size_t)nb * 4096, Xn);
    mm64(SI, Xc, t1);
    mm64(t1, SI, M);
    float* L = log64(M, t1, t2, t3, red);
    for (int e = threadIdx.x; e < 4096; e += TPB) ACC[pidx(e)] += L[pidx(e)];
    __syncthreads();
    float* tmp = Xc; Xc = Xn; Xn = tmp;
  }
  for (int e = threadIdx.x; e < 4096; e += TPB)
    tpart[(size_t)blockIdx.x * 4096 + e] = ACC[pidx(e)];
}

// t = sum(tpart)/B ; newmean = s exp(t) s ; mi = newmean^{-1/2}
__global__ __launch_bounds__(TPB) void k_prep2(const float* __restrict__ tpart,
                                               const float* __restrict__ s_in,
                                               float* __restrict__ mi_out,
                                               int nparts, int Bn) {
  extern __shared__ float sm[];
  float* B0 = sm; float* B1 = sm + MSZ; float* B2 = sm + 2 * MSZ; float* B3 = sm + 3 * MSZ;
  float* red = sm + 4 * MSZ;
  float invB = 1.0f / (float)Bn;
  for (int e = threadIdx.x; e < 4096; e += TPB) {
    float a = 0.f;
    for (int c = 0; c < nparts; ++c) a += tpart[(size_t)c * 4096 + e];
    B0[pidx(e)] = a * invB;
  }
  __syncthreads();
  float* Q = exp64(B0, B1, B2, 1.0f);  // Q = exp(t) in {B1,B2}
  float* F = (Q == B1) ? B2 : B1;
  for (int e = threadIdx.x; e < 4096; e += TPB) B3[pidx(e)] = s_in[e];
  __syncthreads();
  mm64(B3, Q, B0);   // B0 = s @ exp(t)
  mm64(B0, B3, F);   // F  = s exp(t) s = new mean
  float *Y, *Z;
  sqrt_both(F, B0, B3, Q, red, &Y, &Z);
  for (int e = threadIdx.x; e < 4096; e += TPB) mi_out[e] = Z[pidx(e)];
}

// L_b = log(mi x mi) -> Lout (staged in d_out); varpart[blk] = sum ||L_b||_F^2
__global__ __launch_bounds__(TPB) void k_pass2(const float* __restrict__ x,
                                               const float* __restrict__ mi,
                                               float* __restrict__ Lout,
                                               float* __restrict__ varpart, int Bn) {
  extern __shared__ float sm[];
  float* MI = sm;           float* X0 = sm + MSZ;     float* X1 = sm + 2 * MSZ;
  float* M  = sm + 3 * MSZ; float* t1 = sm + 4 * MSZ; float* t2 = sm + 5 * MSZ;
  float* t3 = sm + 6 * MSZ; float* red = sm + 7 * MSZ;
  for (int e = threadIdx.x; e < 4096; e += TPB) MI[pidx(e)] = mi[e];
  int b0 = (int)blockIdx.x;
  if (b0 < Bn) async_load_mat(x + (size_t)b0 * 4096, X0);
  float* Xc = X0; float* Xn = X1;
  float vsum = 0.f;
#pragma unroll 1
  for (int b = b0; b < Bn; b += (int)gridDim.x) {
    wait_async();
    __syncthreads();
    int nb = b + (int)gridDim.x;
    if (nb < Bn) async_load_mat(x + (size_t)nb * 4096, Xn);
    mm64(MI, Xc, t1);
    mm64(t1, MI, M);
    float* L = log64(M, t1, t2, t3, red);
    float part = 0.f;
    for (int e = threadIdx.x; e < 4096; e += TPB) {
      float v = L[pidx(e)];
      part += v * v;
      Lout[(size_t)b * 4096 + e] = v;
    }
    vsum += block_reduce_sum(part, red);  // barriers inside protect L reuse
    float* tmp = Xc; Xc = Xn; Xn = tmp;
  }
  if (threadIdx.x == 0) varpart[blockIdx.x] = vsum;
}

__global__ void k_var(const float* __restrict__ varpart, const float* __restrict__ scale,
                      float* __restrict__ p_out, int n, int Bn) {
  __shared__ float red[256];
  int t = threadIdx.x;
  red[t] = (t < n) ? varpart[t] : 0.0f;
  __syncthreads();
  for (int k = 128; k > 0; k >>= 1) {
    if (t < k) red[t] += red[t + k];
    __syncthreads();
  }
  if (t == 0) {
    float var = red[0] / (float)Bn;
    p_out[0] = scale[0] / (sqrtf(var) + 1e-5f);
  }
}

// out = exp(p * L) in place on d_out; double-buffered async input
__global__ __launch_bounds__(TPB) void k_pass3(float* __restrict__ io,
                                               const float* __restrict__ p_in, int Bn) {
  extern __shared__ float sm[];
  float* T0 = sm; float* T1 = sm + MSZ; float* Bq = sm + 2 * MSZ; float* Bw = sm + 3 * MSZ;
  float p = p_in[0];
  int b0 = (int)blockIdx.x;
  if (b0 < Bn) async_load_mat(io + (size_t)b0 * 4096, T0);
  float* Tc = T0; float* Tn = T1;
#pragma unroll 1
  for (int b = b0; b < Bn; b += (int)gridDim.x) {
    wait_async();
    __syncthreads();
    int nb = b + (int)gridDim.x;
    if (nb < Bn) async_load_mat(io + (size_t)nb * 4096, Tn);
    float* Q = exp64(Tc, Bq, Bw, p);
    for (int e = threadIdx.x; e < 4096; e += TPB) io[(size_t)b * 4096 + e] = Q[pidx(e)];
    __syncthreads();
    float* tmp = Tc; Tc = Tn; Tn = tmp;
  }
}

extern "C" void kernel_launch(void* const* d_in, const int* in_sizes, int n_in,
                              void* d_out, int out_size, void* d_ws, size_t ws_size,
                              hipStream_t stream) {
  (void)n_in; (void)out_size; (void)ws_size;
  const float* x     = (const float*)d_in[0];
  const float* scale = (const float*)d_in[1];
  float* out = (float*)d_out;
  float* ws  = (float*)d_ws;

  const int Bn = in_sizes[0] / 4096;   // 8192
  const int NCHUNK = 32;
  const int GRID = 256;

  float* meanpart = ws;                                  // 32*4096
  float* meanm    = meanpart + (size_t)NCHUNK * 4096;    // 4096
  float* s_m      = meanm + 4096;                        // 4096
  float* si_m     = s_m + 4096;                          // 4096
  float* mi_m     = si_m + 4096;                         // 4096
  float* tpart    = mi_m + 4096;                         // 256*4096
  float* varpart  = tpart + (size_t)GRID * 4096;         // 256
  float* p_s      = varpart + GRID;                      // 1

  size_t sh_prep1 = (size_t)(4 * MSZ + TPB) * sizeof(float);
  size_t sh_pass1 = (size_t)(8 * MSZ + TPB) * sizeof(float);
  size_t sh_prep2 = (size_t)(4 * MSZ + TPB) * sizeof(float);
  size_t sh_pass2 = (size_t)(7 * MSZ + TPB) * sizeof(float);
  size_t sh_pass3 = (size_t)(4 * MSZ) * sizeof(float);

  k_mean_part<<<dim3(16 * NCHUNK), dim3(256), 0, stream>>>(x, meanpart, Bn, NCHUNK);
  k_mean_final<<<dim3(16), dim3(256), 0, stream>>>(meanpart, meanm, Bn, NCHUNK);
  k_prep1<<<dim3(1), dim3(TPB), sh_prep1, stream>>>(meanm, s_m, si_m);
  k_pass1<<<dim3(GRID), dim3(TPB), sh_pass1, stream>>>(x, si_m, tpart, Bn);
  k_prep2<<<dim3(1), dim3(TPB), sh_prep2, stream>>>(tpart, s_m, mi_m, GRID, Bn);
  k_pass2<<<dim3(GRID), dim3(TPB), sh_pass2, stream>>>(x, mi_m, out, varpart, Bn);
  k_var<<<dim3(1), dim3(256), 0, stream>>>(varpart, scale, p_s, GRID, Bn);
  k_pass3<<<dim3(GRID), dim3(TPB), sh_pass3, stream>>>(out, p_s, Bn);
}